// CRF_38585986187871
// MI455X (gfx1250) — compile-verified
//
#include <hip/hip_runtime.h>
#include <stdint.h>

// CRF Viterbi decode, MI455X (gfx1250, wave32).
// (max,+) semiring -> WMMA inapplicable; VALU-bound kernel with CDNA5
// async global->LDS staging (global_load_async_to_lds_b128 / s_wait_asynccnt)
// double-buffering the emissions stream.
//
// Round-1 change: j-blocked (8-wide) max-plus reduction to cut live argmax
// state from 111 regs (sc+m+mi all 37-wide) to ~90, plus a launch-bounds
// occupancy floor of 6 waves/SIMD (round 0 build used ~250 VGPRs -> 4 waves).

#define SEQ  70
#define NT   37
#define BLK  128      // 4 waves of 32; one batch row per lane
#define TROW 40       // transitions row padded to 160B so 8-wide j-blocks are 16B-aligned
#define JB   8        // j-block width (5 blocks: 8,8,8,8,5)

__global__ __launch_bounds__(BLK, 6) void crf_viterbi_kernel(
    const float* __restrict__ em,          // (SEQ, B, NT)
    const unsigned char* __restrict__ mask,// (SEQ, B) bool
    const float* __restrict__ startT,      // (NT)
    const float* __restrict__ endT,        // (NT)
    const float* __restrict__ trans,       // (NT, NT)
    float* __restrict__ out,               // (B, SEQ)
    unsigned char* __restrict__ hist,      // ws: (SEQ-1, NT, B) u8
    int B)
{
    __shared__ __align__(16) float sEm[2][BLK * NT];  // 2 x 18944 B
    __shared__ __align__(16) float sT[NT * TROW];     // 5920 B, padded rows

    const int tid = threadIdx.x;
    const int b   = blockIdx.x * BLK + tid;

    // Stage transitions into LDS with padded rows (one-time 5.5KB, plain loads).
    for (int k = tid; k < NT * NT; k += BLK) {
        sT[(k / NT) * TROW + (k % NT)] = trans[k];
    }

    // Async-stage one timestep of emissions for this block into LDS buffer.
    // Block chunk = 128*37 f32 = 18944 B, contiguous & 16B-aligned in global.
    auto stage = [&](int t, int buf) {
        const float* src = em + (size_t)t * B * NT + (size_t)blockIdx.x * (BLK * NT);
        uint32_t lbase = (uint32_t)(uintptr_t)(&sEm[buf][0]); // LDS byte offset
        const int n128 = (BLK * NT) / 4;                      // 1184 x b128
        for (int k = tid; k < n128; k += BLK) {
            uint32_t dst = lbase + (uint32_t)k * 16u;
            uint64_t g   = (uint64_t)(uintptr_t)(src + (size_t)k * 4);
            asm volatile("global_load_async_to_lds_b128 %0, %1, off"
                         :: "v"(dst), "v"(g) : "memory");
        }
    };

    stage(0, 0);
    asm volatile("s_wait_asynccnt 0" ::: "memory");
    __syncthreads();
    stage(1, 1);   // prefetch t=1 while computing score0

    // score0 = start_transitions + emissions[0]
    float sc[NT];
    #pragma unroll
    for (int j = 0; j < NT; ++j)
        sc[j] = startT[j] + sEm[0][tid * NT + j];

    int mcount = (mask[b] != 0) ? 1 : 0;

    #pragma unroll 1
    for (int t = 1; t < SEQ; ++t) {
        const int cur = t & 1;
        asm volatile("s_wait_asynccnt 0" ::: "memory"); // buffer `cur` (for t) done
        __syncthreads();                                 // visible to all waves
        if (t + 1 < SEQ) stage(t + 1, cur ^ 1);          // overlap DMA w/ compute

        unsigned char* hb = hist + (size_t)(t - 1) * NT * (size_t)B + b;
        float ns[NT];

        // max-plus 37x37, j-blocked: only 8 (m,mi) pairs live at a time.
        #pragma unroll
        for (int j0 = 0; j0 < NT; j0 += JB) {
            float m[JB]; int mi[JB];
            #pragma unroll
            for (int jj = 0; jj < JB; ++jj) {
                if (j0 + jj < NT) { m[jj] = sc[0] + sT[j0 + jj]; mi[jj] = 0; }
            }
            #pragma unroll
            for (int i = 1; i < NT; ++i) {
                const float si = sc[i];
                #pragma unroll
                for (int jj = 0; jj < JB; ++jj) {
                    if (j0 + jj < NT) {
                        float v  = si + sT[i * TROW + j0 + jj];
                        bool  gt = v > m[jj];     // strict > : first-occurrence argmax
                        m[jj]  = gt ? v : m[jj];
                        mi[jj] = gt ? i : mi[jj];
                    }
                }
            }
            #pragma unroll
            for (int jj = 0; jj < JB; ++jj) {
                if (j0 + jj < NT) {
                    // history plane-major: hist[(t-1)][j][b] -> coalesced 1B/lane
                    hb[(size_t)(j0 + jj) * B] = (unsigned char)mi[jj];
                    ns[j0 + jj] = m[jj] + sEm[cur][tid * NT + j0 + jj];
                }
            }
        }

        const bool mk = (mask[(size_t)t * B + b] != 0);
        mcount += mk ? 1 : 0;
        #pragma unroll
        for (int j = 0; j < NT; ++j)
            sc[j] = mk ? ns[j] : sc[j];        // where(mask, best, score)
    }

    // final: + end_transitions, argmax (first occurrence)
    float best = sc[0] + endT[0]; int bj = 0;
    #pragma unroll
    for (int j = 1; j < NT; ++j) {
        float v = sc[j] + endT[j];
        bool gt = v > best;
        best = gt ? v : best;
        bj   = gt ? j : bj;
    }

    const int seq_end = mcount - 1;
    float* ob = out + (size_t)b * SEQ;
    ob[SEQ - 1] = ((SEQ - 1) <= seq_end) ? (float)bj : -1.0f;

    // backtrack (same thread re-reads its own history; same-wave st->ld ordered)
    int curt = bj;
    #pragma unroll 1
    for (int i = SEQ - 1; i >= 1; --i) {
        int prev = hist[((size_t)(i - 1) * NT + curt) * (size_t)B + b];
        curt = (i <= seq_end) ? prev : curt;
        ob[i - 1] = ((i - 1) <= seq_end) ? (float)curt : -1.0f;
    }
}

extern "C" void kernel_launch(void* const* d_in, const int* in_sizes, int n_in,
                              void* d_out, int out_size, void* d_ws, size_t ws_size,
                              hipStream_t stream) {
    const float*         em   = (const float*)d_in[0];
    const unsigned char* mask = (const unsigned char*)d_in[1];
    const float*         st   = (const float*)d_in[2];
    const float*         en   = (const float*)d_in[3];
    const float*         tr   = (const float*)d_in[4];
    float*               outp = (float*)d_out;
    unsigned char*       hist = (unsigned char*)d_ws;   // needs (SEQ-1)*NT*B ~ 84 MB

    const int B = in_sizes[1] / SEQ;   // mask is (SEQ, B)

    dim3 grid(B / BLK), block(BLK);
    crf_viterbi_kernel<<<grid, block, 0, stream>>>(em, mask, st, en, tr, outp, hist, B);
}